// feature_analysis_86011015070114
// MI455X (gfx1250) — compile-verified
//
#include <hip/hip_runtime.h>
#include <hip/hip_bf16.h>
#include <math.h>

// ---------------------------------------------------------------------------
// Problem constants (from reference)
// ---------------------------------------------------------------------------
#define NB    512    // batch rows
#define LLEN  256    // row length
#define FM    64     // pooled feature map (256/4)
#define KBIG  4096   // FM*FM flattened einsum K for ratio pickers
#define HEAD  64
#define EMB   48

typedef __attribute__((ext_vector_type(2))) float v2f;
typedef __attribute__((ext_vector_type(8))) float v8f;

__device__ __forceinline__ float elu1(float v) { return v > 0.0f ? v : (expf(v) - 1.0f); }

// ---------------------------------------------------------------------------
// K1: per-row preprocess: d = diff(x, prepend), smooth = width-19 edge-padded
// box filter, 4-pooled means of x and smoothed diff. One block per row.
// ---------------------------------------------------------------------------
__global__ __launch_bounds__(LLEN)
void preprocess_kernel(const float* __restrict__ x,
                       float* __restrict__ DIFF,   // [NB, 256] smoothed diff
                       float* __restrict__ PX,     // [NB, 64]
                       float* __restrict__ PD)     // [NB, 64]
{
    __shared__ float xs[LLEN];
    __shared__ float db[LLEN];
    __shared__ float sm[LLEN];
    const int b = blockIdx.x;
    const int j = threadIdx.x;

    const float xv = x[b * LLEN + j];
    xs[j] = xv;
    __syncthreads();
    db[j] = (j == 0) ? 0.0f : (xv - xs[j - 1]);
    __syncthreads();

    float s = 0.0f;
#pragma unroll
    for (int t = -9; t <= 9; ++t) {
        int idx = j + t;
        idx = idx < 0 ? 0 : (idx > LLEN - 1 ? LLEN - 1 : idx);
        s += db[idx];
    }
    const float df = s * (1.0f / 19.0f);
    sm[j] = df;
    DIFF[b * LLEN + j] = df;
    __syncthreads();

    if (j < FM) {
        PX[b * FM + j] = 0.25f * (xs[4 * j] + xs[4 * j + 1] + xs[4 * j + 2] + xs[4 * j + 3]);
        PD[b * FM + j] = 0.25f * (sm[4 * j] + sm[4 * j + 1] + sm[4 * j + 2] + sm[4 * j + 3]);
    }
}

// ---------------------------------------------------------------------------
// K0: separable-picker weight reduction:
//   weff[h][K] = sum_I wm[h][I][K] - sum_J wm[h][K][J]
// (because pool(x[:,None,:]-x[:,:,None])[I,J] = px[J]-px[I])
// ---------------------------------------------------------------------------
__global__ __launch_bounds__(256)
void weff_kernel(const float* __restrict__ wm, float* __restrict__ weff)
{
    const int idx = blockIdx.x * 256 + threadIdx.x;   // 0..4095  (h*64 + K)
    const int h  = idx >> 6;
    const int Kc = idx & 63;
    const float* base = wm + h * (FM * FM);
    float s1 = 0.0f, s2 = 0.0f;
    for (int I = 0; I < FM; ++I) s1 += base[I * FM + Kc];
    for (int J = 0; J < FM; ++J) s2 += base[Kc * FM + J];
    weff[idx] = s1 - s2;
}

// ---------------------------------------------------------------------------
// K2: pooled ratio matrices (the only place the 256x256 pairwise structure
// matters):  R[b, I*64+J] = (1/16) * sum_{a,c in 0..3}
//                 (v[4J+c]-u[4I+a]) / (v[4J+c]+u[4I+a]+eps)
// computed for u=v=x (NDI) and u=v=smoothed diff (NDID) in one pass.
// grid (16, NB), 256 threads; rows staged in LDS.
// ---------------------------------------------------------------------------
__global__ __launch_bounds__(256)
void ratio_kernel(const float* __restrict__ x, const float* __restrict__ dsm,
                  float* __restrict__ R1, float* __restrict__ R2)
{
    __shared__ float xr[LLEN];
    __shared__ float dr[LLEN];
    const int b   = blockIdx.y;
    const int tid = threadIdx.x;
    xr[tid] = x[b * LLEN + tid];
    dr[tid] = dsm[b * LLEN + tid];
    __syncthreads();

    const int flat = blockIdx.x * 256 + tid;  // I*64 + J
    const int I = flat >> 6;
    const int J = flat & 63;
    const float eps = 1e-5f;
    float s1 = 0.0f, s2 = 0.0f;
#pragma unroll
    for (int a = 0; a < 4; ++a) {
        const float u1 = xr[4 * I + a];
        const float u2 = dr[4 * I + a];
#pragma unroll
        for (int c = 0; c < 4; ++c) {
            const float v1 = xr[4 * J + c];
            const float v2 = dr[4 * J + c];
            s1 += (v1 - u1) / (v1 + u1 + eps);
            s2 += (v2 - u2) / (v2 + u2 + eps);
        }
    }
    R1[(size_t)b * KBIG + flat] = s1 * (1.0f / 16.0f);
    R2[(size_t)b * KBIG + flat] = s2 * (1.0f / 16.0f);
}

// ---------------------------------------------------------------------------
// K3: one feature picker head, fully on V_WMMA_F32_16X16X4_F32.
//   stage 1:  H[b,h]  = sum_k A[b,k]*W[h,k] + bias[h]    (K = 64 or 4096)
//   stage 2:  E[b,n]  = elu( sum_h elu(H)[b,h]*fw[n,h] + fb[n] )
// Block = 128 threads (4 wave32). Each block owns 16 batch rows.
// Wave w computes head columns 16w..16w+15 in stage 1; waves 0..2 compute the
// 48 emb columns in stage 2 from the LDS-staged elu(H) tile.
//
// f32 WMMA operand layout (16x4 A / 4x16 B, 2 VGPRs each):
//   lanes 0-15  hold K = 4s+0 (v0) and 4s+1 (v1)
//   lanes 16-31 hold K = 4s+2 (v0) and 4s+3 (v1)
// -> per lane, one float2 load per 4-wide K step.
// ---------------------------------------------------------------------------
__global__ __launch_bounds__(128)
void gemm_head_kernel(const float* __restrict__ A,    // [NB, K]
                      const float* __restrict__ W,    // [HEAD, K] row-major
                      const float* __restrict__ bias, // [HEAD]
                      const float* __restrict__ fw,   // [EMB, HEAD] row-major
                      const float* __restrict__ fb,   // [EMB]
                      float* __restrict__ E,          // [NB, EMB]
                      int K)
{
    __shared__ float hmat[16 * HEAD];

    const int tid   = threadIdx.x;
    const int wave  = tid >> 5;
    const int lane  = tid & 31;
    const int laneM = lane & 15;
    const int half  = lane >> 4;       // 0: lanes 0-15, 1: lanes 16-31
    const int b0    = blockIdx.x * 16;

    // ---- stage 1: H tile [16 x 64], wave w covers heads 16w..16w+15 ----
    const int n1 = wave * 16 + laneM;
    const float* arow = A + (size_t)(b0 + laneM) * K;
    const float* wrow = W + (size_t)n1 * K;

    v8f acc = {};
    for (int kb = 0; kb < K; kb += 16) {
#pragma unroll
        for (int s = 0; s < 4; ++s) {
            const int ko = kb + 4 * s + 2 * half;
            v2f a = *(const v2f*)(arow + ko);
            v2f b = *(const v2f*)(wrow + ko);
            acc = __builtin_amdgcn_wmma_f32_16x16x4_f32(false, a, false, b,
                                                        (short)0, acc, false, false);
        }
    }

    const float bv = bias[n1];
#pragma unroll
    for (int r = 0; r < 8; ++r) {
        const int m = r + 8 * half;            // C/D layout: VGPR r -> M=r / r+8
        hmat[m * HEAD + n1] = elu1(acc[r] + bv);
    }
    __syncthreads();

    // ---- stage 2: E tile [16 x 48] = elu(H) x fw^T, waves 0..2 ----
    if (wave < 3) {                             // wave-uniform: EXEC stays all-1s
        const int n2 = wave * 16 + laneM;
        const float* frow = fw + (size_t)n2 * HEAD;
        v8f acc2 = {};
#pragma unroll
        for (int kb = 0; kb < HEAD; kb += 16) {
#pragma unroll
            for (int s = 0; s < 4; ++s) {
                const int ko = kb + 4 * s + 2 * half;
                v2f a = *(const v2f*)(&hmat[laneM * HEAD + ko]);
                v2f b = *(const v2f*)(frow + ko);
                acc2 = __builtin_amdgcn_wmma_f32_16x16x4_f32(false, a, false, b,
                                                             (short)0, acc2, false, false);
            }
        }
        const float fbv = fb[n2];
#pragma unroll
        for (int r = 0; r < 8; ++r) {
            const int m = r + 8 * half;
            E[(size_t)(b0 + m) * EMB + n2] = elu1(acc2[r] + fbv);
        }
    }
}

// ---------------------------------------------------------------------------
// K4: weighted mean over the 4 pickers + fc1(elu) + fc2. One thread per row.
// ---------------------------------------------------------------------------
__global__ __launch_bounds__(256)
void combine_kernel(const float* __restrict__ E0, const float* __restrict__ E1,
                    const float* __restrict__ E2, const float* __restrict__ E3,
                    const float* __restrict__ weight,
                    const float* __restrict__ fc1w, const float* __restrict__ fc1b,
                    const float* __restrict__ fc2w, const float* __restrict__ fc2b,
                    float* __restrict__ out)
{
    const int b = blockIdx.x * 256 + threadIdx.x;
    if (b >= NB) return;
    const float w0 = weight[0], w1 = weight[1], w2 = weight[2], w3 = weight[3];

    float emb[EMB];
#pragma unroll
    for (int j = 0; j < EMB; ++j)
        emb[j] = 0.25f * (w0 * E0[b * EMB + j] + w1 * E1[b * EMB + j] +
                          w2 * E2[b * EMB + j] + w3 * E3[b * EMB + j]);

    float t[32];
#pragma unroll
    for (int i = 0; i < 32; ++i) {
        float s = fc1b[i];
#pragma unroll
        for (int j = 0; j < EMB; ++j) s += fc1w[i * EMB + j] * emb[j];
        t[i] = elu1(s);
    }
#pragma unroll
    for (int o = 0; o < 20; ++o) {
        float s = fc2b[o];
#pragma unroll
        for (int i = 0; i < 32; ++i) s += fc2w[o * 32 + i] * t[i];
        out[b * 20 + o] = s;
    }
}

// ---------------------------------------------------------------------------
// launch
// ---------------------------------------------------------------------------
extern "C" void kernel_launch(void* const* d_in, const int* in_sizes, int n_in,
                              void* d_out, int out_size, void* d_ws, size_t ws_size,
                              hipStream_t stream) {
    const float* x   = (const float*)d_in[0];
    const float* wm0 = (const float*)d_in[1];  const float* bias0 = (const float*)d_in[2];
    const float* fw0 = (const float*)d_in[3];  const float* fb0   = (const float*)d_in[4];
    const float* wm1 = (const float*)d_in[5];  const float* bias1 = (const float*)d_in[6];
    const float* fw1 = (const float*)d_in[7];  const float* fb1   = (const float*)d_in[8];
    const float* wm2 = (const float*)d_in[9];  const float* bias2 = (const float*)d_in[10];
    const float* fw2 = (const float*)d_in[11]; const float* fb2   = (const float*)d_in[12];
    const float* wm3 = (const float*)d_in[13]; const float* bias3 = (const float*)d_in[14];
    const float* fw3 = (const float*)d_in[15]; const float* fb3   = (const float*)d_in[16];
    const float* weight = (const float*)d_in[17];
    const float* fc1w = (const float*)d_in[18]; const float* fc1b = (const float*)d_in[19];
    const float* fc2w = (const float*)d_in[20]; const float* fc2b = (const float*)d_in[21];
    float* out = (float*)d_out;

    // workspace layout (floats); total ~4.50M floats (~17.2 MB)
    float* ws   = (float*)d_ws;
    float* DIFF = ws;                       // NB*256
    float* PX   = DIFF + NB * LLEN;         // NB*64
    float* PD   = PX   + NB * FM;           // NB*64
    float* R1   = PD   + NB * FM;           // NB*4096
    float* R2   = R1   + (size_t)NB * KBIG; // NB*4096
    float* WE0  = R2   + (size_t)NB * KBIG; // 4096
    float* WE2  = WE0  + HEAD * FM;         // 4096
    float* E0   = WE2  + HEAD * FM;         // NB*48
    float* E1   = E0   + NB * EMB;
    float* E2   = E1   + NB * EMB;
    float* E3   = E2   + NB * EMB;

    preprocess_kernel<<<NB, LLEN, 0, stream>>>(x, DIFF, PX, PD);
    weff_kernel<<<16, 256, 0, stream>>>(wm0, WE0);
    weff_kernel<<<16, 256, 0, stream>>>(wm2, WE2);
    ratio_kernel<<<dim3(16, NB), 256, 0, stream>>>(x, DIFF, R1, R2);

    gemm_head_kernel<<<NB / 16, 128, 0, stream>>>(PX, WE0, bias0, fw0, fb0, E0, FM);
    gemm_head_kernel<<<NB / 16, 128, 0, stream>>>(R1, wm1, bias1, fw1, fb1, E1, KBIG);
    gemm_head_kernel<<<NB / 16, 128, 0, stream>>>(PD, WE2, bias2, fw2, fb2, E2, FM);
    gemm_head_kernel<<<NB / 16, 128, 0, stream>>>(R2, wm3, bias3, fw3, fb3, E3, KBIG);

    combine_kernel<<<(NB + 255) / 256, 256, 0, stream>>>(E0, E1, E2, E3, weight,
                                                         fc1w, fc1b, fc2w, fc2b, out);
}